// CustomAttention_71880572666597
// MI455X (gfx1250) — compile-verified
//
#include <hip/hip_runtime.h>
#include <math.h>

// Problem constants (match reference): B=16, R=128, D=128, BD=64
#define B_  16
#define R_  128
#define D_  128
#define BD_ 64

typedef __attribute__((ext_vector_type(2))) float v2f;
typedef __attribute__((ext_vector_type(8))) float v8f;

// One wave32 instruction moves 32 lanes x 16B = 512B (one Wq row) to LDS.
// ASYNCcnt-tracked; completes out-of-band of VGPRs.
#define ASYNC_ROW_TO_LDS(ldsoff_u32, gptr)                                   \
    asm volatile("global_load_async_to_lds_b128 %0, %1, off"                 \
                 :: "v"(ldsoff_u32), "v"(gptr) : "memory")

#define WAIT_ASYNCCNT(n)                                                     \
    asm volatile("s_wait_asynccnt " #n ::: "memory")

// ---------------------------------------------------------------------------
// Kernel 1: K = kv @ Wk + bk ; V = kv @ Wv + bv ; Vn = V / max(||V||, 1e-12)
// grid = B*R rows, block = 128 threads
// ---------------------------------------------------------------------------
__global__ __launch_bounds__(128) void proj_kv_kernel(
    const float* __restrict__ kv, const float* __restrict__ Wk,
    const float* __restrict__ bk, const float* __restrict__ Wv,
    const float* __restrict__ bv,
    float* __restrict__ Kmat, float* __restrict__ Vn)
{
    __shared__ float sx[D_];
    __shared__ float sv[BD_];
    const int row = blockIdx.x;          // flattened (b, r)
    const int tid = threadIdx.x;

    sx[tid] = kv[row * D_ + tid];
    __syncthreads();

    float kacc = bk[tid];
#pragma unroll 8
    for (int i = 0; i < D_; ++i)
        kacc = fmaf(sx[i], Wk[i * D_ + tid], kacc);
    Kmat[row * D_ + tid] = kacc;

    float vacc = 0.0f;
    if (tid < BD_) {
        vacc = bv[tid];
#pragma unroll 8
        for (int i = 0; i < D_; ++i)
            vacc = fmaf(sx[i], Wv[i * BD_ + tid], vacc);
        sv[tid] = vacc;
    }
    __syncthreads();

    if (tid < BD_) {
        float ss = 0.0f;
#pragma unroll 8
        for (int j = 0; j < BD_; ++j)
            ss = fmaf(sv[j], sv[j], ss);
        const float nrm = fmaxf(sqrtf(ss), 1e-12f);
        Vn[row * BD_ + tid] = vacc / nrm;
    }
}

// ---------------------------------------------------------------------------
// Kernel 2 (dominant; streams the 1 GiB Wq exactly once):
//   For each (s,t): Qtile = q_embed[:,s,:] (16x128) @ Wq[s,t] (128x128)
//   aw_raw[b,s,t] = relu( sum_d (Qtile[b,d] + bq[s,t,d]) * K[b,t,d] )
//
// One 256-thread block (8 wave32) per (s,t); wave w owns d-cols [16w,16w+16).
// Wq tile is staged global->LDS with GLOBAL_LOAD_ASYNC_TO_LDS_B128 in 4
// stages of 32 rows, double-buffered (ASYNCcnt + barrier pipelining). Each
// wave copies 4 rows/stage with one async b128 per row (512B contiguous).
// LDS rows padded to 132 dwords so the two lane-halves of fragment reads hit
// disjoint banks. K=128 runs as 32 V_WMMA_F32_16X16X4_F32 split over two
// interleaved accumulators to halve the RAW chain on the matrix pipe.
//
// Fragment layouts (CDNA5 ISA 7.12.2, 32-bit operands, wave32):
//   A 16x4 : v0 = A[M=lane&15][K = 0|2 by lane-half], v1 = K+1
//   B 4x16 : v0 = B[K = 0|2 by lane-half][N=lane&15], v1 = K+1
//   C 16x16: vj = C[M = j + 8*(lane>=16)][N = lane&15]
// ---------------------------------------------------------------------------
#define SW_STRIDE 132   // dwords per staged Wq row (128 + 4 pad)

__global__ __launch_bounds__(256) void score_wmma_kernel(
    const float* __restrict__ q_embed, const float* __restrict__ Wq,
    const float* __restrict__ bq, const float* __restrict__ Kmat,
    float* __restrict__ aw_raw)
{
    __shared__ float sA[D_][B_ + 1];          // q_embed[:,s,:]^T, +1 pad
    __shared__ float sW[2][32 * SW_STRIDE];   // double-buffered Wq stage
    __shared__ float s_part[8][B_];           // per-wave score partials

    const int t   = blockIdx.x;
    const int s   = blockIdx.y;
    const int tid = threadIdx.x;
    const int lane = tid & 31;
    const int w    = tid >> 5;                // wave 0..7 -> d-cols [16w,16w+16)

    // Stage A = q_embed[:, s, :] (16 x 128) into LDS, transposed.
    for (int idx = tid; idx < B_ * D_; idx += 256) {
        const int b = idx >> 7;               // D_ == 128
        const int i = idx & (D_ - 1);
        sA[i][b] = q_embed[(b * R_ + s) * D_ + i];
    }

    const float* __restrict__ wq = Wq + ((size_t)(s * R_ + t)) * (D_ * D_);
    const int half   = lane >> 4;             // lane half 0|1
    const int l15    = lane & 15;
    const int rowOff = half << 1;             // K sub-row 0|2
    const int col    = (w << 4) + l15;        // output column d

    // Byte offset of sW within the workgroup LDS allocation (low 32 bits of
    // the generic address of a __shared__ object are the LDS byte address).
    const uint32_t swBase = (uint32_t)(uintptr_t)(&sW[0][0]);
    const uint32_t laneByte = (uint32_t)(lane << 4);   // 16B per lane

    // Issue stage 0: wave w copies rows 4w..4w+3 into buffer 0.
#pragma unroll
    for (int i = 0; i < 4; ++i) {
        const int r = (w << 2) + i;
        const float* g = wq + (size_t)r * D_ + (lane << 2);
        const uint32_t l = swBase + (uint32_t)(r * SW_STRIDE * 4) + laneByte;
        ASYNC_ROW_TO_LDS(l, g);
    }

    v8f acc0 = {0.f, 0.f, 0.f, 0.f, 0.f, 0.f, 0.f, 0.f};
    v8f acc1 = {0.f, 0.f, 0.f, 0.f, 0.f, 0.f, 0.f, 0.f};

    for (int st = 0; st < 4; ++st) {
        if (st < 3) {   // prefetch next 32-row stage into the other buffer
            const int buf = (st + 1) & 1;
#pragma unroll
            for (int i = 0; i < 4; ++i) {
                const int r  = (w << 2) + i;                 // row within stage
                const int gr = ((st + 1) << 5) + r;          // global Wq row
                const float* g = wq + (size_t)gr * D_ + (lane << 2);
                const uint32_t l = swBase +
                    (uint32_t)((buf * 32 + r) * SW_STRIDE * 4) + laneByte;
                ASYNC_ROW_TO_LDS(l, g);
            }
            WAIT_ASYNCCNT(0x4);   // this stage's 4 rows have landed
        } else {
            WAIT_ASYNCCNT(0x0);   // final stage fully landed
        }
        __syncthreads();          // all waves' rows of this stage visible
                                  // (also covers the sA staging before st 0)

        const float* bufp = &sW[st & 1][0];
        const int kbase = st << 5;
#pragma unroll
        for (int c = 0; c < 8; ++c) {
            const int k0 = c << 2;
            v2f a, b;
            a.x = sA[kbase + k0 + rowOff][l15];
            a.y = sA[kbase + k0 + rowOff + 1][l15];
            b.x = bufp[(k0 + rowOff) * SW_STRIDE + col];
            b.y = bufp[(k0 + rowOff + 1) * SW_STRIDE + col];
            if (c & 1)
                acc1 = __builtin_amdgcn_wmma_f32_16x16x4_f32(
                           false, a, false, b, (short)0, acc1, false, false);
            else
                acc0 = __builtin_amdgcn_wmma_f32_16x16x4_f32(
                           false, a, false, b, (short)0, acc0, false, false);
        }
        __syncthreads();          // done reading buffer before it is reused
    }

    // Fused score over this wave's 16 d-columns, reduced per batch row.
    const float bqv  = bq[(s * R_ + t) * D_ + col];
    const int   Moff = half << 3;
#pragma unroll
    for (int j = 0; j < 8; ++j) {
        const int bb = j + Moff;
        float c = (acc0[j] + acc1[j] + bqv) * Kmat[(bb * R_ + t) * D_ + col];
        c += __shfl_xor(c, 1, 32);
        c += __shfl_xor(c, 2, 32);
        c += __shfl_xor(c, 4, 32);
        c += __shfl_xor(c, 8, 32);
        if (l15 == 0) s_part[w][bb] = c;
    }
    __syncthreads();

    if (tid < B_) {
        float sum = 0.0f;
#pragma unroll
        for (int ww = 0; ww < 8; ++ww) sum += s_part[ww][tid];
        aw_raw[(tid * R_ + s) * R_ + t] = fmaxf(sum, 0.0f);   // relu
    }
}

// ---------------------------------------------------------------------------
// Kernel 3: sum-normalize scores, emit aw; attended = aw @ Vn ; out = @Wr + br
// grid = B*R (one (b,s) row), block = 128 threads
// ---------------------------------------------------------------------------
__global__ __launch_bounds__(128) void finalize_kernel(
    const float* __restrict__ aw_raw, const float* __restrict__ Vn,
    const float* __restrict__ Wr, const float* __restrict__ br,
    float* __restrict__ attended_out, float* __restrict__ aw_out)
{
    __shared__ float sred[128];
    __shared__ float sawn[R_];
    __shared__ float satt[BD_];

    const int bs  = blockIdx.x;        // b*R + s
    const int tid = threadIdx.x;

    const float r = aw_raw[bs * R_ + tid];
    sred[tid] = r;
    __syncthreads();
    for (int off = 64; off > 0; off >>= 1) {   // deterministic tree reduce
        if (tid < off) sred[tid] += sred[tid + off];
        __syncthreads();
    }
    const float total = sred[0];
    const float awn = r / (1e-8f + total);
    aw_out[bs * R_ + tid] = awn;
    sawn[tid] = awn;
    __syncthreads();

    if (tid < BD_) {
        const int b = bs >> 7;         // R_ == 128
        float att = 0.0f;
#pragma unroll 8
        for (int tt = 0; tt < R_; ++tt)
            att = fmaf(sawn[tt], Vn[(b * R_ + tt) * BD_ + tid], att);
        satt[tid] = att;
    }
    __syncthreads();

    float o = br[tid];
#pragma unroll 8
    for (int j = 0; j < BD_; ++j)
        o = fmaf(satt[j], Wr[j * D_ + tid], o);
    attended_out[bs * D_ + tid] = o;
}

// ---------------------------------------------------------------------------
extern "C" void kernel_launch(void* const* d_in, const int* in_sizes, int n_in,
                              void* d_out, int out_size, void* d_ws, size_t ws_size,
                              hipStream_t stream)
{
    (void)in_sizes; (void)n_in; (void)out_size; (void)ws_size;

    const float* q_embed = (const float*)d_in[0];
    const float* kv      = (const float*)d_in[1];
    const float* Wk      = (const float*)d_in[2];
    const float* bk      = (const float*)d_in[3];
    const float* Wv      = (const float*)d_in[4];
    const float* bv      = (const float*)d_in[5];
    const float* Wq      = (const float*)d_in[6];
    const float* bq      = (const float*)d_in[7];
    const float* Wr      = (const float*)d_in[8];
    const float* br      = (const float*)d_in[9];

    // workspace layout (all fully rewritten every call)
    float* Kmat   = (float*)d_ws;                         // B*R*D  = 262144
    float* Vn     = Kmat + (size_t)B_ * R_ * D_;          // B*R*BD = 131072
    float* aw_raw = Vn   + (size_t)B_ * R_ * BD_;         // B*R*R  = 262144

    float* attended_out = (float*)d_out;                  // (B,R,D) first
    float* aw_out = attended_out + (size_t)B_ * R_ * D_;  // then (B,R,R)

    proj_kv_kernel<<<B_ * R_, 128, 0, stream>>>(kv, Wk, bk, Wv, bv, Kmat, Vn);

    dim3 grid_st(R_, R_);   // x = t, y = s
    score_wmma_kernel<<<grid_st, 256, 0, stream>>>(q_embed, Wq, bq, Kmat, aw_raw);

    finalize_kernel<<<B_ * R_, 128, 0, stream>>>(aw_raw, Vn, Wr, br,
                                                 attended_out, aw_out);
}